// IDWFeatureInterpolator_90383291777517
// MI455X (gfx1250) — compile-verified
//
#include <hip/hip_runtime.h>

typedef __attribute__((ext_vector_type(2))) float v2f;
typedef __attribute__((ext_vector_type(8))) float v8f;

#define NQ   8192
#define NS   4096
#define FD   256
#define KNN  8
#define WAVES 8      // waves per block (256 threads, wave32)
#define QPW  16      // queries per wave (one WMMA N-tile)

// Branchless sorted insert into ascending top-K (dk[KNN-1] is current worst).
__device__ __forceinline__ void topk_insert(float (&dk)[KNN], int (&ik)[KNN],
                                            float d, int i) {
  if (d >= dk[KNN - 1]) return;
  float dj = d; int ij = i;
#pragma unroll
  for (int j = 0; j < KNN; ++j) {
    bool sm = dj < dk[j];
    float td = dk[j]; int ti = ik[j];
    dk[j] = sm ? dj : dk[j];
    ik[j] = sm ? ij : ik[j];
    dj = sm ? td : dj;
    ij = sm ? ti : ij;
  }
}

__global__ __launch_bounds__(256)
void idw_topk_wmma_kernel(const float* __restrict__ qc,   // (B, NQ, 3)
                          const float* __restrict__ sc,   // (NS, 3)
                          const float* __restrict__ sf,   // (B, NS, FD)
                          float* __restrict__ out) {      // (B, NQ, FD)
  // Sensor data in A-matrix form: (-2x, -2y, -2z, |s|^2)
  __shared__ float4 sA[NS];   // 64 KB (WGP has 320 KB LDS)

  const int tid  = threadIdx.x;
  const int lane = tid & 31;
  const int wave = tid >> 5;
  const int b    = blockIdx.y;
  const int qBase = (blockIdx.x * WAVES + wave) * QPW;

  for (int s = tid; s < NS; s += 256) {
    float x = sc[s * 3 + 0], y = sc[s * 3 + 1], z = sc[s * 3 + 2];
    sA[s] = make_float4(-2.f * x, -2.f * y, -2.f * z, x * x + y * y + z * z);
  }
  __syncthreads();

  // ---- B matrix (queries): N = lane%16 holds query qBase+N, K-col [qx,qy,qz,1]
  const int  n  = lane & 15;
  const bool hi = lane >= 16;
  const float* q = qc + ((size_t)b * NQ + (qBase + n)) * 3;
  const float qx = q[0], qy = q[1], qz = q[2];
  const float q2 = qx * qx + qy * qy + qz * qz;
  v2f bmat;
  bmat.x = hi ? qz : qx;      // lanes 0-15: K=0 | lanes 16-31: K=2
  bmat.y = hi ? 1.0f : qy;    // lanes 0-15: K=1 | lanes 16-31: K=3

  float dk[KNN]; int ik[KNN];
#pragma unroll
  for (int j = 0; j < KNN; ++j) { dk[j] = 3.4e38f; ik[j] = 0; }

  // ---- main loop: 16 sensors per WMMA tile; lane holds 8 d^2 for its query.
  // Unroll x2 so two independent WMMA + ds_load chains overlap.
#pragma unroll 2
  for (int m0 = 0; m0 < NS; m0 += 16) {
    float4 s4 = sA[m0 + n];   // sensor row M = lane%16 of this tile
    v2f amat;
    amat.x = hi ? s4.z : s4.x;   // K=2(-2z) | K=0(-2x)
    amat.y = hi ? s4.w : s4.y;   // K=3(s^2) | K=1(-2y)

    v8f c;
#pragma unroll
    for (int r = 0; r < 8; ++r) c[r] = q2;   // seed accumulator with |q|^2

    // d2[m, n] = |q|^2 + |s|^2 - 2 q.s   (computed entirely by the WMMA)
    c = __builtin_amdgcn_wmma_f32_16x16x4_f32(false, amat, false, bmat,
                                              (short)0, c, false, false);

    // Single guard for the whole tile: min of the 8 candidates vs current worst.
    float cmin = fminf(fminf(fminf(c[0], c[1]), fminf(c[2], c[3])),
                       fminf(fminf(c[4], c[5]), fminf(c[6], c[7])));
    if (cmin < dk[KNN - 1]) {
#pragma unroll
      for (int r = 0; r < 8; ++r) {
        // C layout: VGPR r, lanes 0-15 -> M=r ; lanes 16-31 -> M=r+8
        int si = m0 + r + (hi ? 8 : 0);
        topk_insert(dk, ik, c[r], si);
      }
    }
  }

  // ---- merge the two sensor halves: lane L <-> lane L^16 (same query)
  float pd[KNN]; int pi[KNN];
#pragma unroll
  for (int j = 0; j < KNN; ++j) {
    pd[j] = __shfl_xor(dk[j], 16, 32);
    pi[j] = __shfl_xor(ik[j], 16, 32);
  }
#pragma unroll
  for (int j = 0; j < KNN; ++j) topk_insert(dk, ik, pd[j], pi[j]);

  // ---- IDW weights (matches reference: w = 1/(sqrt(max(d2,0)) + eps), normalized)
  float w[KNN]; float wsum = 0.f;
#pragma unroll
  for (int j = 0; j < KNN; ++j) {
    w[j] = 1.0f / (sqrtf(fmaxf(dk[j], 0.0f)) + 1e-8f);
    wsum += w[j];
  }
  const float winv = 1.0f / wsum;
#pragma unroll
  for (int j = 0; j < KNN; ++j) w[j] *= winv;

  // ---- phase 2: cooperative feature gather + weighted sum.
  // Lane covers 8 consecutive feature channels: f = lane*8 .. lane*8+7.
  const float* sfb = sf + (size_t)b * NS * FD;
  for (int qq = 0; qq < QPW; ++qq) {
    float4 a0 = make_float4(0.f, 0.f, 0.f, 0.f);
    float4 a1 = make_float4(0.f, 0.f, 0.f, 0.f);
#pragma unroll
    for (int j = 0; j < KNN; ++j) {
      float wj = __shfl(w[j], qq, 32);
      int   sj = __shfl(ik[j], qq, 32);
      const float4* fp = (const float4*)(sfb + (size_t)sj * FD + lane * 8);
      float4 f0 = fp[0], f1 = fp[1];
      a0.x = fmaf(wj, f0.x, a0.x); a0.y = fmaf(wj, f0.y, a0.y);
      a0.z = fmaf(wj, f0.z, a0.z); a0.w = fmaf(wj, f0.w, a0.w);
      a1.x = fmaf(wj, f1.x, a1.x); a1.y = fmaf(wj, f1.y, a1.y);
      a1.z = fmaf(wj, f1.z, a1.z); a1.w = fmaf(wj, f1.w, a1.w);
    }
    float4* op = (float4*)(out + ((size_t)b * NQ + (qBase + qq)) * FD + lane * 8);
    op[0] = a0;
    op[1] = a1;
  }
}

extern "C" void kernel_launch(void* const* d_in, const int* in_sizes, int n_in,
                              void* d_out, int out_size, void* d_ws, size_t ws_size,
                              hipStream_t stream) {
  const float* qc = (const float*)d_in[0];   // query_coords  (B, NQ, 3)
  const float* sc = (const float*)d_in[1];   // sensor_coords (NS, 3)
  const float* sf = (const float*)d_in[2];   // sensor_features (B, NS, FD)
  float* out = (float*)d_out;                // (B, NQ, FD)

  const int Bn = in_sizes[0] / (NQ * 3);
  dim3 grid(NQ / (WAVES * QPW), Bn);
  idw_topk_wmma_kernel<<<grid, 256, 0, stream>>>(qc, sc, sf, out);
}